// AttnDecoderRNN_27084063769023
// MI455X (gfx1250) — compile-verified
//
#include <hip/hip_runtime.h>
#include <hip/hip_bf16.h>
#include <math.h>

typedef float v2f __attribute__((ext_vector_type(2)));
typedef float v8f __attribute__((ext_vector_type(8)));

#define Hdim 1024
#define Vdim 50257
#define Ldim 256

// ---------------------------------------------------------------------------
// WMMA fp32 GEMV: y[M] = act( W0[M,K] @ xs0 (+ W1[M,K] @ xs1) + b0 (+ b1) )
// One wave handles 16 rows via V_WMMA_F32_16X16X4_F32, x vector(s) in LDS.
// A-tile layout (ISA 7.12.2, 32-bit A 16x4): lanes 0-15 hold (K=k0,k0+1) of
// row=lane; lanes 16-31 hold (K=k0+2,k0+3) of row=lane-16  -> one b64 load.
// B column 0 = x[k0..k0+3]; other columns of D are unused garbage.
// D column 0: lane 0 VGPR j = row rowBase+j, lane 16 VGPR j = row rowBase+8+j.
// NT=true marks the weight stream non-temporal (use when W >> L2 = 192MB).
// ---------------------------------------------------------------------------
template <bool NT>
__device__ __forceinline__ v2f load_w(const float* p)
{
    if (NT) return __builtin_nontemporal_load((const v2f*)p);
    return *(const v2f*)p;
}

template <bool NT>
__global__ __launch_bounds__(256)
void gemv_wmma(const float* __restrict__ W0, const float* __restrict__ x0,
               const float* __restrict__ W1, const float* __restrict__ x1,
               const float* __restrict__ b0, const float* __restrict__ b1,
               float* __restrict__ y, int M, int K, int relu)
{
    __shared__ float xs[2048];                 // K<=2048 single, 2*K<=2048 dual
    for (int i = threadIdx.x; i < K; i += blockDim.x) xs[i] = x0[i];
    if (W1 != nullptr)
        for (int i = threadIdx.x; i < K; i += blockDim.x) xs[K + i] = x1[i];
    __syncthreads();

    const int lane = threadIdx.x & 31;
    const int wave = threadIdx.x >> 5;
    const int half = lane >> 4;                // 0 -> K offs {0,1}, 1 -> {2,3}
    const int r    = lane & 15;
    const int rowBase = (blockIdx.x * (blockDim.x >> 5) + wave) * 16;

    int rowL = rowBase + r;                    // clamped row for safe loads
    if (rowL > M - 1) rowL = M - 1;

    v8f c = {0.f, 0.f, 0.f, 0.f, 0.f, 0.f, 0.f, 0.f};

    const float* wp0 = W0 + (size_t)rowL * K + 2 * half;
    #pragma unroll 16
    for (int k0 = 0; k0 < K; k0 += 4) {
        v2f a = load_w<NT>(wp0 + k0);
        v2f b; b.x = xs[k0 + 2 * half]; b.y = xs[k0 + 2 * half + 1];
        c = __builtin_amdgcn_wmma_f32_16x16x4_f32(false, a, false, b,
                                                  (short)0, c, false, false);
    }
    if (W1 != nullptr) {
        const float* wp1 = W1 + (size_t)rowL * K + 2 * half;
        const float* xb  = xs + K;
        #pragma unroll 16
        for (int k0 = 0; k0 < K; k0 += 4) {
            v2f a = load_w<NT>(wp1 + k0);
            v2f b; b.x = xb[k0 + 2 * half]; b.y = xb[k0 + 2 * half + 1];
            c = __builtin_amdgcn_wmma_f32_16x16x4_f32(false, a, false, b,
                                                      (short)0, c, false, false);
        }
    }

    if (r == 0) {                              // lanes 0 and 16 hold column 0
        const int base = rowBase + half * 8;
        #pragma unroll
        for (int j = 0; j < 8; ++j) {
            int row = base + j;
            if (row < M) {
                float v = c[j];
                if (b0) v += b0[row];
                if (b1) v += b1[row];
                if (relu) v = fmaxf(v, 0.f);
                y[row] = v;
            }
        }
    }
}

// ---------------------------------------------------------------------------
// embedded = emb[feat]; attn_in = [embedded | h0]; combined[0:H] = embedded
// ---------------------------------------------------------------------------
__global__ void prep_kernel(const int* __restrict__ feat,
                            const float* __restrict__ emb,
                            const float* __restrict__ hidden,
                            float* __restrict__ attn_in,
                            float* __restrict__ combined)
{
    int i = blockIdx.x * blockDim.x + threadIdx.x;
    if (i < Hdim) {
        float e = emb[(size_t)feat[0] * Hdim + i];
        attn_in[i]  = e;
        combined[i] = e;
    } else if (i < 2 * Hdim) {
        attn_in[i] = hidden[i - Hdim];
    }
}

// softmax over 256 scores; writes to ws copy and to output section
__global__ void softmax_kernel(const float* __restrict__ scores,
                               float* __restrict__ w_ws,
                               float* __restrict__ w_out)
{
    __shared__ float red[256];
    int tid = threadIdx.x;
    float v = scores[tid];
    red[tid] = v; __syncthreads();
    for (int s = 128; s > 0; s >>= 1) {
        if (tid < s) red[tid] = fmaxf(red[tid], red[tid + s]);
        __syncthreads();
    }
    float m = red[0]; __syncthreads();
    float e = expf(v - m);
    red[tid] = e; __syncthreads();
    for (int s = 128; s > 0; s >>= 1) {
        if (tid < s) red[tid] += red[tid + s];
        __syncthreads();
    }
    float p = e / red[0];
    w_ws[tid]  = p;
    w_out[tid] = p;
}

// attn_applied[j] = sum_l w[l] * enc[l*H + j]   (coalesced column streaming)
__global__ void attn_apply_kernel(const float* __restrict__ w,
                                  const float* __restrict__ enc,
                                  float* __restrict__ dst)
{
    int j = blockIdx.x * blockDim.x + threadIdx.x;   // 0..H-1
    float acc = 0.f;
    #pragma unroll 8
    for (int l = 0; l < Ldim; ++l) acc += w[l] * enc[l * Hdim + j];
    dst[j] = acc;
}

__device__ __forceinline__ float sigm(float x) { return 1.f / (1.f + expf(-x)); }

// PyTorch gate order i,f,g,o on a [4H] gate vector
__global__ void lstm_elem_kernel(const float* __restrict__ g,
                                 const float* __restrict__ c_prev,
                                 float* __restrict__ h_out,
                                 float* __restrict__ c_out)
{
    int i = blockIdx.x * blockDim.x + threadIdx.x;   // 0..H-1
    float ig = sigm(g[i]);
    float fg = sigm(g[Hdim + i]);
    float gg = tanhf(g[2 * Hdim + i]);
    float og = sigm(g[3 * Hdim + i]);
    float c  = fg * c_prev[i] + ig * gg;
    c_out[i] = c;
    h_out[i] = og * tanhf(c);
}

// logsumexp stats over the V logits
__global__ void lse_kernel(const float* __restrict__ logits, int n,
                           float* __restrict__ mls)
{
    __shared__ float red[1024];
    int tid = threadIdx.x;
    float m = -3.402823466e38f;
    for (int i = tid; i < n; i += 1024) m = fmaxf(m, logits[i]);
    red[tid] = m; __syncthreads();
    for (int s = 512; s > 0; s >>= 1) {
        if (tid < s) red[tid] = fmaxf(red[tid], red[tid + s]);
        __syncthreads();
    }
    m = red[0]; __syncthreads();
    float sum = 0.f;
    for (int i = tid; i < n; i += 1024) sum += expf(logits[i] - m);
    red[tid] = sum; __syncthreads();
    for (int s = 512; s > 0; s >>= 1) {
        if (tid < s) red[tid] += red[tid + s];
        __syncthreads();
    }
    if (tid == 0) { mls[0] = m; mls[1] = logf(red[0]); }
}

__global__ void lse_apply_kernel(float* __restrict__ logp, int n,
                                 const float* __restrict__ mls)
{
    int i = blockIdx.x * blockDim.x + threadIdx.x;
    if (i < n) logp[i] = logp[i] - mls[0] - mls[1];
}

// ---------------------------------------------------------------------------
extern "C" void kernel_launch(void* const* d_in, const int* in_sizes, int n_in,
                              void* d_out, int out_size, void* d_ws, size_t ws_size,
                              hipStream_t stream)
{
    (void)in_sizes; (void)n_in; (void)out_size; (void)ws_size;

    const int*   feat   = (const int*)  d_in[0];
    const float* hidden = (const float*)d_in[1];   // [2*H]
    const float* cell   = (const float*)d_in[2];   // [2*H]
    const float* enc    = (const float*)d_in[3];   // [L,H]
    const float* emb    = (const float*)d_in[4];   // [V,H]
    const float* W_attn = (const float*)d_in[5];   // [L,2H]
    const float* b_attn = (const float*)d_in[6];
    const float* W_comb = (const float*)d_in[7];   // [H,2H]
    const float* b_comb = (const float*)d_in[8];
    const float* W_out  = (const float*)d_in[9];   // [V,H]
    const float* b_out  = (const float*)d_in[10];
    const float* w_ih[2] = {(const float*)d_in[11], (const float*)d_in[15]};
    const float* w_hh[2] = {(const float*)d_in[12], (const float*)d_in[16]};
    const float* b_ih[2] = {(const float*)d_in[13], (const float*)d_in[17]};
    const float* b_hh[2] = {(const float*)d_in[14], (const float*)d_in[18]};

    float* out = (float*)d_out;
    float* ws  = (float*)d_ws;
    // workspace layout (floats)
    float* attn_in  = ws;             // 2048
    float* scores   = ws + 2048;      // 256
    float* weights  = ws + 2304;      // 256
    float* combined = ws + 2560;      // 2048
    float* xrelu    = ws + 4608;      // 1024
    float* gates    = ws + 5632;      // 4096
    float* mls      = ws + 9728;      // 2
    // output layout: [logprobs V | h_out 2H | c_out 2H | attn_weights L]
    float* logp   = out;
    float* h_out  = out + Vdim;
    float* c_out  = out + Vdim + 2 * Hdim;
    float* aw_out = out + Vdim + 4 * Hdim;

    // 1) embed gather + concat
    prep_kernel<<<8, 256, 0, stream>>>(feat, emb, hidden, attn_in, combined);

    // 2) attention scores: [256 x 2048] GEMV  (16 tiles -> 2 blocks of 8 waves)
    gemv_wmma<false><<<2, 256, 0, stream>>>(W_attn, attn_in, nullptr, nullptr,
                                            b_attn, nullptr, scores, Ldim, 2 * Hdim, 0);

    // 3) softmax over L (also emits attn_weights output)
    softmax_kernel<<<1, 256, 0, stream>>>(scores, weights, aw_out);

    // 4) attn_applied = w @ enc  -> combined[H:2H]
    attn_apply_kernel<<<Hdim / 256, 256, 0, stream>>>(weights, enc, combined + Hdim);

    // 5) x = relu([1024 x 2048] GEMV)   (64 tiles -> 8 blocks)
    gemv_wmma<false><<<8, 256, 0, stream>>>(W_comb, combined, nullptr, nullptr,
                                            b_comb, nullptr, xrelu, Hdim, 2 * Hdim, 1);

    // 6) LSTM layer 0: gates = W_ih@x + W_hh@h0 + biases  (256 tiles -> 32 blocks)
    gemv_wmma<false><<<32, 256, 0, stream>>>(w_ih[0], xrelu, w_hh[0], hidden,
                                             b_ih[0], b_hh[0], gates, 4 * Hdim, Hdim, 0);
    lstm_elem_kernel<<<Hdim / 256, 256, 0, stream>>>(gates, cell, h_out, c_out);

    // 7) LSTM layer 1 (x = h0_new)
    gemv_wmma<false><<<32, 256, 0, stream>>>(w_ih[1], h_out, w_hh[1], hidden + Hdim,
                                             b_ih[1], b_hh[1], gates, 4 * Hdim, Hdim, 0);
    lstm_elem_kernel<<<Hdim / 256, 256, 0, stream>>>(gates, cell + Hdim,
                                                     h_out + Hdim, c_out + Hdim);

    // 8) logits = [50257 x 1024] GEMV. 206 MB > 192 MB L2 -> non-temporal
    //    weight stream so the L2-resident LSTM/attn weights survive replays.
    const int tiles  = (Vdim + 15) / 16;        // 3142
    const int blocks = (tiles + 7) / 8;         // 393
    gemv_wmma<true><<<blocks, 256, 0, stream>>>(W_out, h_out + Hdim, nullptr, nullptr,
                                                b_out, nullptr, logp, Vdim, Hdim, 0);

    // 9) log_softmax
    lse_kernel<<<1, 1024, 0, stream>>>(logp, Vdim, mls);
    lse_apply_kernel<<<(Vdim + 255) / 256, 256, 0, stream>>>(logp, Vdim, mls);
}